// IDMMD_40355512713519
// MI455X (gfx1250) — compile-verified
//
#include <hip/hip_runtime.h>

// Problem constants (fixed by the reference).
#define NCLS  1024
#define DIMS  256
#define NROWS 131072
#define NTOT  2048          // 2*NCLS stacked centers
#define NTILE 128           // NTOT/16 tiles per side

typedef __attribute__((ext_vector_type(2))) float v2f;
typedef __attribute__((ext_vector_type(8))) float v8f;

// Workspace layout (float offsets)
// acc[0]=dist sum, acc[1]=loss sum, acc[2..6]=precomputed exp2 scales
#define OFF_COUNTS 0
#define OFF_SUMR   (OFF_COUNTS + NCLS)
#define OFF_SUMI   (OFF_SUMR + NCLS*DIMS)
#define OFF_ACC    (OFF_SUMI + NCLS*DIMS)
#define OFF_TOTAL  (OFF_ACC + 8)
#define OFF_SQ     (OFF_TOTAL + NTOT*DIMS)
#define ZERO_N     OFF_TOTAL                     // zero counts+sums+acc

__device__ __forceinline__ float fast_exp2(float x) {
    return __builtin_amdgcn_exp2f(x);            // v_exp_f32
}

// ---------------------------------------------------------------- zero
__global__ void idmmd_zero(float* ws, int n) {
    for (int i = blockIdx.x * blockDim.x + threadIdx.x; i < n;
         i += gridDim.x * blockDim.x)
        ws[i] = 0.0f;
}

// ------------------------------------------------------- segment scatter
// One thread per (row, 4-dim chunk): float4 load + 4 native f32 atomics,
// for both modalities. Lane with q==0 also bumps the class count.
__global__ void idmmd_scatter(const float* __restrict__ xr,
                              const float* __restrict__ xi,
                              const int*   __restrict__ tgt,
                              float* __restrict__ ws) {
    int idx = blockIdx.x * blockDim.x + threadIdx.x;   // 0 .. NROWS*64
    int n = idx >> 6;
    int q = idx & 63;
    int c = tgt[n];
    float4 vr = ((const float4*)xr)[idx];
    float4 vi = ((const float4*)xi)[idx];
    float* sr = ws + OFF_SUMR + c * DIMS + q * 4;
    float* si = ws + OFF_SUMI + c * DIMS + q * 4;
    unsafeAtomicAdd(sr + 0, vr.x);
    unsafeAtomicAdd(sr + 1, vr.y);
    unsafeAtomicAdd(sr + 2, vr.z);
    unsafeAtomicAdd(sr + 3, vr.w);
    unsafeAtomicAdd(si + 0, vi.x);
    unsafeAtomicAdd(si + 1, vi.y);
    unsafeAtomicAdd(si + 2, vi.z);
    unsafeAtomicAdd(si + 3, vi.w);
    if (q == 0) unsafeAtomicAdd(ws + OFF_COUNTS + c, 1.0f);
}

// ------------------------------------------------- centers + row norms
// One block per center row (2048 blocks x 256 threads).
__global__ void idmmd_centers(float* __restrict__ ws) {
    int b = blockIdx.x;           // 0..2047
    int d = threadIdx.x;          // 0..255
    int cls = b & (NCLS - 1);
    float denom = fmaxf(ws[OFF_COUNTS + cls], 1.0f);
    const float* sums = ws + (b < NCLS ? OFF_SUMR : OFF_SUMI);
    float v = sums[cls * DIMS + d] / denom;
    ws[OFF_TOTAL + b * DIMS + d] = v;

    __shared__ float red[256];
    red[d] = v * v;
    __syncthreads();
    for (int s = 128; s > 0; s >>= 1) {
        if (d < s) red[d] += red[d + s];
        __syncthreads();
    }
    if (d == 0) ws[OFF_SQ + b] = red[0];
}

// ---------------------------------------------------- WMMA Gram passes
// One wave -> one 16x16 tile of total @ total^T via V_WMMA_F32_16X16X4_F32.
// pass==1: accumulate sum(dists) into acc[0].
// pass==2: accumulate signed 5-term Gaussian kernel sum into acc[1],
//          using precomputed scales acc[2+i] = log2(e)/(bw*2^i) so each
//          term is one multiply + one v_exp_f32 (no IEEE divides).
__global__ void idmmd_gram(const float* __restrict__ T,
                           const float* __restrict__ sq,
                           float* __restrict__ acc, int pass) {
    const int waveInBlk = threadIdx.x >> 5;
    const int wave  = blockIdx.x * (blockDim.x >> 5) + waveInBlk;
    const int lane  = threadIdx.x & 31;
    const int hl    = lane >> 4;          // half of wave: selects K pair
    const int lrow  = lane & 15;          // M (for A) / N (for B) index
    const int tileI = wave >> 7;          // wave / NTILE
    const int tileJ = wave & (NTILE - 1);

    // A lane pattern: lanes 0-15 -> (M=lrow, K=k0,k0+1); lanes 16-31 -> (M=lrow, K=k0+2,k0+3)
    // For Gram (B = T^T of the J tile) the B operand uses the identical pattern with base tileJ.
    const float* arow = T + (tileI * 16 + lrow) * DIMS + 2 * hl;
    const float* brow = T + (tileJ * 16 + lrow) * DIMS + 2 * hl;

    v8f c = {};
    #pragma unroll 8
    for (int k = 0; k < DIMS; k += 4) {
        v2f a = *(const v2f*)(arow + k);
        v2f b = *(const v2f*)(brow + k);
        c = __builtin_amdgcn_wmma_f32_16x16x4_f32(
                /*neg_a=*/false, a, /*neg_b=*/false, b,
                /*c_mod=*/(short)0, c, /*reuse_a=*/false, /*reuse_b=*/false);
    }

    // C/D layout: lanes 0-15 VGPR r -> (M=r, N=lrow); lanes 16-31 -> (M=r+8, N=lrow)
    const float sqj = sq[tileJ * 16 + lrow];
    // Tiles are 16-aligned and 1024 % 16 == 0 -> block sign uniform per tile.
    const float sign = ((tileI < NTILE / 2) == (tileJ < NTILE / 2)) ? 1.0f : -1.0f;

    float sc0 = 0.f, sc1 = 0.f, sc2 = 0.f, sc3 = 0.f, sc4 = 0.f;
    if (pass == 2) {   // uniform scalar loads of precomputed exp2 scales
        sc0 = acc[2]; sc1 = acc[3]; sc2 = acc[4]; sc3 = acc[5]; sc4 = acc[6];
    }

    float local = 0.0f;
    #pragma unroll
    for (int r = 0; r < 8; ++r) {
        int m  = r + hl * 8;
        int gi = tileI * 16 + m;
        float d = sq[gi] + sqj - 2.0f * c[r];
        d = fmaxf(d, 0.0f);
        if (pass == 1) {
            local += d;
        } else {
            float kern = fast_exp2(-d * sc0) + fast_exp2(-d * sc1)
                       + fast_exp2(-d * sc2) + fast_exp2(-d * sc3)
                       + fast_exp2(-d * sc4);
            local += sign * kern;
        }
    }

    // wave32 reduce, then one atomic per block
    for (int off = 16; off > 0; off >>= 1)
        local += __shfl_down(local, off, 32);
    __shared__ float part[8];
    if (lane == 0) part[waveInBlk] = local;
    __syncthreads();
    if (threadIdx.x == 0) {
        float s = 0.0f;
        #pragma unroll
        for (int w = 0; w < 8; ++w) s += part[w];
        unsafeAtomicAdd(&acc[pass == 1 ? 0 : 1], s);
    }
}

// ------------------------------------------------------------ bandwidth
// bandwidth = sum(dists)/(n^2-n) / KERNEL_MUL^(KERNEL_NUM//2) = /4
// Precompute acc[2+i] = log2(e) / (bw * 2^i) so pass 2 uses exp2(-d*scale).
__global__ void idmmd_bw(float* __restrict__ acc) {
    const float n = (float)NTOT;
    float bw = acc[0] / (n * n - n) / 4.0f;
    float s  = 1.4426950408889634f / bw;   // log2(e)/bw
    #pragma unroll
    for (int i = 0; i < 5; ++i) { acc[2 + i] = s; s *= 0.5f; }
}

// ---------------------------------------------------------------- final
__global__ void idmmd_final(const float* __restrict__ acc, float* __restrict__ out) {
    out[0] = acc[1] * (1.0f / ((float)NCLS * (float)NCLS));
}

extern "C" void kernel_launch(void* const* d_in, const int* in_sizes, int n_in,
                              void* d_out, int out_size, void* d_ws, size_t ws_size,
                              hipStream_t stream) {
    const float* xr  = (const float*)d_in[0];
    const float* xi  = (const float*)d_in[1];
    const int*   tgt = (const int*)d_in[2];
    float* ws  = (float*)d_ws;
    float* out = (float*)d_out;
    float* acc = ws + OFF_ACC;

    // 1) zero counts + sums + acc
    idmmd_zero<<<1024, 256, 0, stream>>>(ws, ZERO_N);

    // 2) segment scatter: NROWS * 64 threads
    idmmd_scatter<<<(NROWS * 64) / 256, 256, 0, stream>>>(xr, xi, tgt, ws);

    // 3) centers + squared norms: one block per center row
    idmmd_centers<<<NTOT, 256, 0, stream>>>(ws);

    // 4) pass 1: sum of pairwise distances (WMMA Gram)
    //    128x128 tiles, 8 waves/block -> 2048 blocks, exact cover (EXEC all ones)
    idmmd_gram<<<(NTILE * NTILE) / 8, 256, 0, stream>>>(ws + OFF_TOTAL, ws + OFF_SQ, acc, 1);

    // 5) bandwidth + exp2 scale precompute
    idmmd_bw<<<1, 1, 0, stream>>>(acc);

    // 6) pass 2: signed multi-kernel sum
    idmmd_gram<<<(NTILE * NTILE) / 8, 256, 0, stream>>>(ws + OFF_TOTAL, ws + OFF_SQ, acc, 2);

    // 7) loss = signed sum / (1024*1024)
    idmmd_final<<<1, 1, 0, stream>>>(acc, out);
}